// PE_7249904795814
// MI455X (gfx1250) — compile-verified
//
#include <hip/hip_runtime.h>
#include <cstdint>
#include <cstddef>

// Problem geometry (fixed by the reference).
#define IN_H      512
#define IN_W      512
#define NCH       3
#define NB        32
#define PW        128               // patch size
#define TILE_ROWS 64                // output rows per block
#define SRC_ROWS  (TILE_ROWS/4 + 2) // 18 source rows cover one tile
#define NTHREADS  256               // 8 wave32s

#define AS_GLOBAL __attribute__((address_space(1)))
#define AS_LDS    __attribute__((address_space(3)))

typedef float v4f __attribute__((ext_vector_type(4)));

#if __has_builtin(__builtin_amdgcn_global_load_async_to_lds_b32) && \
    __has_builtin(__builtin_amdgcn_s_wait_asynccnt)
#define USE_ASYNC_LDS 1
#else
#define USE_ASYNC_LDS 0
#endif

__global__ __launch_bounds__(NTHREADS) void crop_resize_x4_kernel(
    const float* __restrict__ in,     // [B, C, H, W]
    const int*   __restrict__ starts, // [B, 2] (i1, i2)
    float*       __restrict__ out)    // [B, C, H, W]
{
    const int tile = blockIdx.x;   // 0..7  (64 output rows each)
    const int ch   = blockIdx.y;   // 0..2
    const int b    = blockIdx.z;   // 0..31
    const int t    = threadIdx.x;

    __shared__ float smem[SRC_ROWS][PW];   // 9 KiB

    const int i1 = starts[2*b + 0];
    const int i2 = starts[2*b + 1];
    const int y0 = tile * TILE_ROWS;
    const int ks = (y0 >> 2) - 1;          // first (unclamped) source row

    const float* src =
        in + ((size_t)(b * NCH + ch) * IN_H + i1) * IN_W + i2;

    // ---- Stage the 18 needed patch rows into LDS (CDNA5 async copy path) ----
    for (int e = t; e < SRC_ROWS * PW; e += NTHREADS) {
        const int j   = e >> 7;        // local row 0..17
        const int col = e & (PW - 1);  // patch column 0..127
        int gr = ks + j;               // clamped source row (edge replication)
        gr = gr < 0 ? 0 : (gr > PW - 1 ? PW - 1 : gr);
        const float* gp = src + (size_t)gr * IN_W + col;
#if USE_ASYNC_LDS
        __builtin_amdgcn_global_load_async_to_lds_b32(
            (AS_GLOBAL int*)gp, (AS_LDS int*)&smem[j][col],
            /*offset=*/0, /*cpol=*/0);
#else
        smem[j][col] = *gp;
#endif
    }
#if USE_ASYNC_LDS
    __builtin_amdgcn_s_wait_asynccnt(0);   // my wave's async LDS writes done
#endif
    __syncthreads();                       // everyone's LDS writes visible

    // ---- Compute: each thread owns one source cell c -> 4 output columns ----
    const int c   = t & (PW - 1);          // 0..127
    const int rp  = t >> 7;                // 0 or 1 (row pair)
    const int cm1 = c > 0 ? c - 1 : 0;
    const int cp1 = c < PW - 1 ? c + 1 : c;

    float* dst = out + ((size_t)(b * NCH + ch) * IN_H + y0) * IN_W;

#pragma unroll 8
    for (int i = 0; i < TILE_ROWS / 2; ++i) {
        const int yl = 2 * i + rp;         // row within tile, 0..63
        const int y  = y0 + yl;
        const int k  = y >> 2;
        const int r  = y & 3;
        // vertical frac: {0.625, 0.875, 0.125, 0.375}[r]
        const float fv = 0.25f * (float)((r + 2) & 3) + 0.125f;
        const int rlo  = k - 1 + (r >> 1); // unclamped floor row
        const int la   = rlo - ks;         // 0..16 (smem rows hold clamped data)
        const int lb   = la + 1;           // <= 17

        const float a0 = smem[la][cm1], a1 = smem[la][c], a2 = smem[la][cp1];
        const float b0 = smem[lb][cm1], b1 = smem[lb][c], b2 = smem[lb][cp1];
        const float g  = 1.0f - fv;
        const float v0 = g * a0 + fv * b0;
        const float v1 = g * a1 + fv * b1;
        const float v2 = g * a2 + fv * b2;

        v4f o;
        o.x = 0.375f * v0 + 0.625f * v1;   // r=0: floor c-1, frac .625
        o.y = 0.125f * v0 + 0.875f * v1;   // r=1: floor c-1, frac .875
        o.z = 0.875f * v1 + 0.125f * v2;   // r=2: floor c,   frac .125
        o.w = 0.625f * v1 + 0.375f * v2;   // r=3: floor c,   frac .375

        v4f* dp = (v4f*)(dst + (size_t)yl * IN_W + (c << 2));
#if __has_builtin(__builtin_nontemporal_store)
        __builtin_nontemporal_store(o, dp);  // stream 96 MiB past L2
#else
        *dp = o;
#endif
    }
}

extern "C" void kernel_launch(void* const* d_in, const int* in_sizes, int n_in,
                              void* d_out, int out_size, void* d_ws, size_t ws_size,
                              hipStream_t stream) {
    const float* in     = (const float*)d_in[0];
    const int*   starts = (const int*)d_in[1];
    float*       out    = (float*)d_out;

    dim3 grid(IN_H / TILE_ROWS, NCH, NB);   // (8, 3, 32) = 768 blocks
    crop_resize_x4_kernel<<<grid, NTHREADS, 0, stream>>>(in, starts, out);
}